// GraphConvolution_67396626808861
// MI455X (gfx1250) — compile-verified
//
#include <hip/hip_runtime.h>
#include <hip/hip_bf16.h>

typedef __attribute__((ext_vector_type(2))) float v2f;
typedef __attribute__((ext_vector_type(8))) float v8f;
typedef __attribute__((ext_vector_type(4))) unsigned int u32x4;
typedef __attribute__((ext_vector_type(8))) int i32x8;
typedef __attribute__((ext_vector_type(4))) int i32x4;

#define N_NODES 100000
#define N_EDGES 1600000
#define IN_DIM  512
#define OUT_DIM 256

// LDS row strides in floats (hardware padding via TDM pad_amount = 4 DWORDs)
#define AS_STRIDE 36    // 32 + 4
#define BS_STRIDE 132   // 128 + 4
#define ABUF_F (64 * AS_STRIDE)            // 2304 floats
#define BBUF_F (32 * BS_STRIDE)            // 4224 floats
#define STAGE_F (ABUF_F + BBUF_F)          // 6528 floats per stage

// ---------------------------------------------------------------------------
// TDM: issue a 2D tensor_load_to_lds. Dims/strides in elements (4B each).
// D# group0: [count=1][lds_addr][global_addr lo][global_addr hi | type=2]
// D# group1: data_size=4B, pad_enable, pad_interval/amount, dims, tiles, stride
// Groups 2/3 (and the trailing vector operand) are zero for a 2D tensor.
// ---------------------------------------------------------------------------
__device__ __forceinline__ void tdm_load_2d(unsigned lds_addr, const float* gaddr,
                                            unsigned tensor_d0, unsigned tensor_d1,
                                            unsigned tile_d0, unsigned tile_d1,
                                            unsigned stride0,
                                            unsigned pad_interval, unsigned pad_amount) {
    unsigned long long ga = (unsigned long long)(uintptr_t)gaddr;
    u32x4 g0;
    g0[0] = 1u;                                            // count=1, no gather
    g0[1] = lds_addr;                                      // LDS byte address
    g0[2] = (unsigned)ga;                                  // global addr [31:0]
    g0[3] = (unsigned)((ga >> 32) & 0x1FFFFFFu) | (2u << 30);  // [56:32] | type=2

    i32x8 g1;
    g1[0] = (int)((2u << 16)                 // data_size = 4 bytes
                | (1u << 20)                 // pad_enable
                | (pad_interval << 22)       // 2^(k+1) DWORDs between pads
                | (pad_amount << 25));       // (k+1) DWORDs of pad
    g1[1] = (int)(tensor_d0 << 16);                          // tensor_dim0[15:0]
    g1[2] = (int)((tensor_d0 >> 16) | (tensor_d1 << 16));    // d0[31:16] | d1[15:0]
    g1[3] = (int)((tensor_d1 >> 16) | (tile_d0 << 16));      // d1[31:16] | tile_d0
    g1[4] = (int)tile_d1;                                    // tile_d1 | tile_d2=0
    g1[5] = (int)stride0;                                    // dim0 stride [31:0]
    g1[6] = 0;
    g1[7] = 0;

    i32x4 z4 = {0, 0, 0, 0};
    i32x8 z8 = {0, 0, 0, 0, 0, 0, 0, 0};
    __builtin_amdgcn_tensor_load_to_lds(g0, g1, z4, z4, z8, 0);
}

// ---------------------------------------------------------------------------
// GEMM: support[N_NODES][256] = x[N_NODES][512] @ W[512][256]
// block: 64(M) x 128(N), 8 waves as 2(M) x 4(N), wave = 32x32 via 4 f32 WMMA
// accumulators. K-tile = 32, double-buffered LDS filled by the TDM.
// ---------------------------------------------------------------------------
__global__ __launch_bounds__(256) void gcn_gemm_wmma(const float* __restrict__ x,
                                                     const float* __restrict__ W,
                                                     float* __restrict__ support) {
    __shared__ float lds[2 * STAGE_F];

    const int m0     = blockIdx.x * 64;
    const int n0     = blockIdx.y * 128;
    const int tid    = threadIdx.x;
    const int waveId = tid >> 5;
    const int lane   = tid & 31;
    const int half   = lane >> 4;   // 0: lanes 0-15, 1: lanes 16-31
    const int lrow   = lane & 15;
    const int wm     = waveId & 1;  // M wave: 0..1
    const int wn     = waveId >> 1; // N wave: 0..3

    const unsigned lds_base = (unsigned)(uintptr_t)&lds[0];

    v8f acc[2][2] = {};

    // Prologue: DMA first K-tile into stage 0 (wave 0 drives the TDM; EXEC-free)
    if (waveId == 0) {
        tdm_load_2d(lds_base, x + (size_t)m0 * IN_DIM,
                    IN_DIM, (unsigned)(N_NODES - m0),   // OOB rows read as zero
                    32, 64, IN_DIM, /*pad: 32 dw*/ 4, /*+4 dw*/ 3);
        tdm_load_2d(lds_base + ABUF_F * 4, W + n0,
                    OUT_DIM, IN_DIM,
                    128, 32, OUT_DIM, /*pad: 128 dw*/ 6, /*+4 dw*/ 3);
    }

    int buf = 0;
    for (int kt = 0; kt < IN_DIM; kt += 32) {
        const int nxt = buf ^ 1;
        if (waveId == 0) {
            if (kt + 32 < IN_DIM) {
                // Prefetch next tile into the other stage, then retire current
                tdm_load_2d(lds_base + nxt * STAGE_F * 4,
                            x + (size_t)m0 * IN_DIM + (kt + 32),
                            (unsigned)(IN_DIM - (kt + 32)), (unsigned)(N_NODES - m0),
                            32, 64, IN_DIM, 4, 3);
                tdm_load_2d(lds_base + (nxt * STAGE_F + ABUF_F) * 4,
                            W + (size_t)(kt + 32) * OUT_DIM + n0,
                            OUT_DIM, (unsigned)(IN_DIM - (kt + 32)),
                            128, 32, OUT_DIM, 6, 3);
                __builtin_amdgcn_s_wait_tensorcnt(2);  // in-order: current tile done
            } else {
                __builtin_amdgcn_s_wait_tensorcnt(0);
            }
        }
        __syncthreads();  // publish current stage to all waves

        const float* As = lds + buf * STAGE_F;
        const float* Bs = lds + buf * STAGE_F + ABUF_F;

        #pragma unroll
        for (int kk = 0; kk < 32; kk += 4) {
            const int kb = kk + 2 * half;  // ISA A/B layout: half-wave K split
            v2f a0, a1, b0, b1;
            a0.x = As[(wm * 32 + lrow) * AS_STRIDE + kb];
            a0.y = As[(wm * 32 + lrow) * AS_STRIDE + kb + 1];
            a1.x = As[(wm * 32 + 16 + lrow) * AS_STRIDE + kb];
            a1.y = As[(wm * 32 + 16 + lrow) * AS_STRIDE + kb + 1];
            b0.x = Bs[kb * BS_STRIDE + wn * 32 + lrow];
            b0.y = Bs[(kb + 1) * BS_STRIDE + wn * 32 + lrow];
            b1.x = Bs[kb * BS_STRIDE + wn * 32 + 16 + lrow];
            b1.y = Bs[(kb + 1) * BS_STRIDE + wn * 32 + 16 + lrow];

            acc[0][0] = __builtin_amdgcn_wmma_f32_16x16x4_f32(false, a0, false, b0,
                            (short)0, acc[0][0], false, false);
            acc[0][1] = __builtin_amdgcn_wmma_f32_16x16x4_f32(false, a0, false, b1,
                            (short)0, acc[0][1], false, false);
            acc[1][0] = __builtin_amdgcn_wmma_f32_16x16x4_f32(false, a1, false, b0,
                            (short)0, acc[1][0], false, false);
            acc[1][1] = __builtin_amdgcn_wmma_f32_16x16x4_f32(false, a1, false, b1,
                            (short)0, acc[1][1], false, false);
        }
        __syncthreads();  // all waves done reading this stage before TDM reuses it
        buf = nxt;
    }

    // Write back: C/D layout — VGPR i holds M=i (lanes 0-15) and M=i+8 (lanes 16-31)
    #pragma unroll
    for (int ms = 0; ms < 2; ++ms) {
        #pragma unroll
        for (int ns = 0; ns < 2; ++ns) {
            const int gc  = n0 + wn * 32 + ns * 16 + lrow;
            const int gr0 = m0 + wm * 32 + ms * 16 + half * 8;
            #pragma unroll
            for (int i = 0; i < 8; ++i) {
                const int gr = gr0 + i;
                if (gr < N_NODES)
                    support[(size_t)gr * OUT_DIM + gc] = acc[ms][ns][i];
            }
        }
    }
}

// ---------------------------------------------------------------------------
// COO SpMM scatter: out[row] += val * support[col]; one wave per edge,
// each lane owns 8 consecutive floats (2x float4 gather, 8 f32 atomics).
// support & out are L2-resident (102 MB each < 192 MB L2).
// ---------------------------------------------------------------------------
__global__ __launch_bounds__(256) void gcn_scatter(const float* __restrict__ support,
                                                   const int* __restrict__ erow,
                                                   const int* __restrict__ ecol,
                                                   const float* __restrict__ eval,
                                                   float* __restrict__ out) {
    const int e = (blockIdx.x << 3) + (threadIdx.x >> 5);
    if (e >= N_EDGES) return;
    const int lane = threadIdx.x & 31;
    const int r = erow[e];
    const int c = ecol[e];
    const float v = eval[e];

    const float4* s = (const float4*)(support + (size_t)c * OUT_DIM) + lane * 2;
    const float4 m0 = s[0];
    const float4 m1 = s[1];

    float* o = out + (size_t)r * OUT_DIM + lane * 8;
    __hip_atomic_fetch_add(o + 0, m0.x * v, __ATOMIC_RELAXED, __HIP_MEMORY_SCOPE_AGENT);
    __hip_atomic_fetch_add(o + 1, m0.y * v, __ATOMIC_RELAXED, __HIP_MEMORY_SCOPE_AGENT);
    __hip_atomic_fetch_add(o + 2, m0.z * v, __ATOMIC_RELAXED, __HIP_MEMORY_SCOPE_AGENT);
    __hip_atomic_fetch_add(o + 3, m0.w * v, __ATOMIC_RELAXED, __HIP_MEMORY_SCOPE_AGENT);
    __hip_atomic_fetch_add(o + 4, m1.x * v, __ATOMIC_RELAXED, __HIP_MEMORY_SCOPE_AGENT);
    __hip_atomic_fetch_add(o + 5, m1.y * v, __ATOMIC_RELAXED, __HIP_MEMORY_SCOPE_AGENT);
    __hip_atomic_fetch_add(o + 6, m1.z * v, __ATOMIC_RELAXED, __HIP_MEMORY_SCOPE_AGENT);
    __hip_atomic_fetch_add(o + 7, m1.w * v, __ATOMIC_RELAXED, __HIP_MEMORY_SCOPE_AGENT);
}

// ---------------------------------------------------------------------------
// Epilogue: out = relu(out + b), N_NODES*256 elements (exact multiple of 256)
// ---------------------------------------------------------------------------
__global__ __launch_bounds__(256) void gcn_bias_relu(float* __restrict__ out,
                                                     const float* __restrict__ b) {
    const size_t i = (size_t)blockIdx.x * 256 + threadIdx.x;
    const float v = out[i] + b[i & (OUT_DIM - 1)];
    out[i] = v > 0.f ? v : 0.f;
}

extern "C" void kernel_launch(void* const* d_in, const int* in_sizes, int n_in,
                              void* d_out, int out_size, void* d_ws, size_t ws_size,
                              hipStream_t stream) {
    const float* x    = (const float*)d_in[0];
    const float* W    = (const float*)d_in[1];
    const float* b    = (const float*)d_in[2];
    const int*   erow = (const int*)d_in[3];
    const int*   ecol = (const int*)d_in[4];
    const float* eval = (const float*)d_in[5];
    float* out     = (float*)d_out;
    float* support = (float*)d_ws;   // N_NODES * OUT_DIM floats (102.4 MB)

    const int total = N_NODES * OUT_DIM;

    // out accumulates via atomics -> must start at zero every call
    (void)hipMemsetAsync(out, 0, (size_t)total * sizeof(float), stream);

    dim3 ggrid((N_NODES + 63) / 64, OUT_DIM / 128);
    gcn_gemm_wmma<<<ggrid, 256, 0, stream>>>(x, W, support);

    gcn_scatter<<<(N_EDGES + 7) / 8, 256, 0, stream>>>(support, erow, ecol, eval, out);

    gcn_bias_relu<<<total / 256, 256, 0, stream>>>(out, b);
}